// GATLayer_51994874085584
// MI455X (gfx1250) — compile-verified
//
#include <hip/hip_runtime.h>
#include <hip/hip_bf16.h>
#include <math.h>

// ---------------------------------------------------------------------------
// GAT layer for MI455X (gfx1250, wave32).
//   N=100000 nodes, E=1.6M edges, IN_DIM=128, HEADS=8, F=16 (out 128).
//   Projection uses V_WMMA_F32_16X16X4_F32 (native f32 WMMA -> exact f32 math).
//   Edge phase: xp (51MB) + out (51MB) are L2-resident (192MB L2), so the
//   random gathers + 205M f32 atomics stay on-chip.
// ---------------------------------------------------------------------------

typedef __attribute__((ext_vector_type(2))) float v2f;
typedef __attribute__((ext_vector_type(8))) float v8f;

#define GAT_IN_DIM   128
#define GAT_HEADS    8
#define GAT_FPH      16
#define GAT_OUT      128
#define NEG_SLOPE    0.2f

// ---------------------------------------------------------------------------
// Kernel 1: xp = x * W   (N x 128) * (128 x 128), f32 WMMA 16x16x4.
// One block = 16 rows of x staged in LDS; 8 waves, wave w owns head w's
// 16-column tile. 32 WMMAs per wave (K=128 in steps of 4).
// A-frag layout (ISA 7.12.2, 32-bit A 16x4): lane l holds M=l&15,
//   vgpr0=K(=2*(l>>4)), vgpr1=K+1.
// B-frag layout (4x16): vgpr i, lanes0-15: K=i, N=lane; lanes16-31: K=i+2.
// C/D layout: vgpr i, lanes0-15: M=i, N=lane; lanes16-31: M=i+8.
// ---------------------------------------------------------------------------
__global__ __launch_bounds__(256) void gat_proj_wmma(
    const float* __restrict__ x, const float* __restrict__ W,
    float* __restrict__ xp, int N) {
  __shared__ float sx[16 * GAT_IN_DIM];  // 8 KB

  const int row0 = blockIdx.x * 16;
  const int tid  = threadIdx.x;

  // Stage 16x128 tile of x into LDS (256 threads * 8 floats).
  for (int i = tid; i < 16 * GAT_IN_DIM; i += 256) {
    int r = i >> 7, c = i & 127;
    int gr = row0 + r;
    sx[i] = (gr < N) ? x[gr * GAT_IN_DIM + c] : 0.0f;
  }
  __syncthreads();

  const int wave = tid >> 5;        // 0..7 == head / 16-col tile
  const int lane = tid & 31;
  const int n    = lane & 15;       // N coordinate within tile
  const int half = lane >> 4;       // 0 or 1
  const int col  = wave * 16 + n;   // global output column
  const int arow = lane & 15;       // M coordinate for A frag

  v8f acc = {};
  for (int k0 = 0; k0 < GAT_IN_DIM; k0 += 4) {
    const int kk = k0 + 2 * half;
    v2f a, b;
    a[0] = sx[arow * GAT_IN_DIM + kk];
    a[1] = sx[arow * GAT_IN_DIM + kk + 1];
    b[0] = W[kk * GAT_OUT + col];
    b[1] = W[(kk + 1) * GAT_OUT + col];
    acc = __builtin_amdgcn_wmma_f32_16x16x4_f32(
        /*neg_a=*/false, a, /*neg_b=*/false, b,
        /*c_mod=*/(short)0, acc, /*reuse_a=*/false, /*reuse_b=*/false);
  }

  // Write back D: vgpr i -> row (half*8 + i), column col.
  const int mbase = half * 8;
#pragma unroll
  for (int i = 0; i < 8; ++i) {
    int gr = row0 + mbase + i;
    if (gr < N) xp[gr * GAT_OUT + col] = acc[i];
  }
}

// ---------------------------------------------------------------------------
// Kernel 2: per-(node, head) attention halves:
//   al_src[n,h] = dot(xp[n, h*16 : h*16+16], a_src[h]);  same for a_dst.
// ---------------------------------------------------------------------------
__global__ void gat_al(const float* __restrict__ xp,
                       const float* __restrict__ a_src,
                       const float* __restrict__ a_dst,
                       float* __restrict__ al_src,
                       float* __restrict__ al_dst, int N) {
  int idx = blockIdx.x * blockDim.x + threadIdx.x;  // node*8 + h
  if (idx >= N * GAT_HEADS) return;
  int h = idx & 7, node = idx >> 3;
  const float* r  = xp + node * GAT_OUT + h * GAT_FPH;
  const float* as = a_src + h * GAT_FPH;
  const float* ad = a_dst + h * GAT_FPH;
  float s = 0.0f, d = 0.0f;
#pragma unroll
  for (int f = 0; f < GAT_FPH; ++f) {
    float v = r[f];
    s += v * as[f];
    d += v * ad[f];
  }
  al_src[idx] = s;
  al_dst[idx] = d;
}

// ---------------------------------------------------------------------------
// Kernel 3: init accumulators (out=0, m=-inf, denom=0).
// ---------------------------------------------------------------------------
__global__ void gat_init(float* __restrict__ out, float* __restrict__ m,
                         float* __restrict__ denom, int N) {
  int i = blockIdx.x * blockDim.x + threadIdx.x;
  if (i < N * GAT_OUT) out[i] = 0.0f;
  if (i < N * GAT_HEADS) {
    m[i] = -__builtin_inff();
    denom[i] = 0.0f;
  }
}

// Float atomic max via signed/unsigned ordering trick (valid incl. -inf init).
__device__ __forceinline__ void atomicMaxF(float* addr, float v) {
  if (v >= 0.0f)
    atomicMax((int*)addr, __float_as_int(v));
  else
    atomicMin((unsigned int*)addr, __float_as_uint(v));
}

__device__ __forceinline__ float leaky(float e) {
  return e >= 0.0f ? e : NEG_SLOPE * e;
}

// ---------------------------------------------------------------------------
// Kernel 4: per-edge segment max:  m[dst,h] = max(m, leakyrelu(al_s + al_d)).
// ---------------------------------------------------------------------------
__global__ void gat_edge_max(const int* __restrict__ src,
                             const int* __restrict__ dst,
                             const float* __restrict__ al_src,
                             const float* __restrict__ al_dst,
                             float* __restrict__ m, int E) {
  int e = blockIdx.x * blockDim.x + threadIdx.x;
  if (e >= E) return;
  int s = src[e], d = dst[e];
  const float4* as = (const float4*)(al_src + s * GAT_HEADS);
  const float4* ad = (const float4*)(al_dst + d * GAT_HEADS);
  float4 a0 = as[0], a1 = as[1], b0 = ad[0], b1 = ad[1];
  float va[8] = {a0.x, a0.y, a0.z, a0.w, a1.x, a1.y, a1.z, a1.w};
  float vb[8] = {b0.x, b0.y, b0.z, b0.w, b1.x, b1.y, b1.z, b1.w};
#pragma unroll
  for (int h = 0; h < GAT_HEADS; ++h)
    atomicMaxF(&m[d * GAT_HEADS + h], leaky(va[h] + vb[h]));
}

// ---------------------------------------------------------------------------
// Kernel 5: per-edge exp-sum:  denom[dst,h] += exp(e - m[dst,h]).
// ---------------------------------------------------------------------------
__global__ void gat_edge_expsum(const int* __restrict__ src,
                                const int* __restrict__ dst,
                                const float* __restrict__ al_src,
                                const float* __restrict__ al_dst,
                                const float* __restrict__ m,
                                float* __restrict__ denom, int E) {
  int e = blockIdx.x * blockDim.x + threadIdx.x;
  if (e >= E) return;
  int s = src[e], d = dst[e];
  const float4* as = (const float4*)(al_src + s * GAT_HEADS);
  const float4* ad = (const float4*)(al_dst + d * GAT_HEADS);
  const float4* mm = (const float4*)(m + d * GAT_HEADS);
  float4 a0 = as[0], a1 = as[1], b0 = ad[0], b1 = ad[1], m0 = mm[0], m1 = mm[1];
  float va[8] = {a0.x, a0.y, a0.z, a0.w, a1.x, a1.y, a1.z, a1.w};
  float vb[8] = {b0.x, b0.y, b0.z, b0.w, b1.x, b1.y, b1.z, b1.w};
  float vm[8] = {m0.x, m0.y, m0.z, m0.w, m1.x, m1.y, m1.z, m1.w};
#pragma unroll
  for (int h = 0; h < GAT_HEADS; ++h)
    atomicAdd(&denom[d * GAT_HEADS + h], expf(leaky(va[h] + vb[h]) - vm[h]));
}

// ---------------------------------------------------------------------------
// Kernel 6: per-edge weighted scatter:  out[dst] += alpha * xp[src].
// One wave32 per edge: lane handles 4 consecutive cols (float4 gather,
// 4 atomic f32 adds). alpha recomputed per head (cheap, L2/L1 cached).
// ---------------------------------------------------------------------------
__global__ __launch_bounds__(256) void gat_edge_agg(
    const int* __restrict__ src, const int* __restrict__ dst,
    const float* __restrict__ al_src, const float* __restrict__ al_dst,
    const float* __restrict__ m, const float* __restrict__ denom,
    const float* __restrict__ xp, float* __restrict__ out, int E) {
  int gwarp  = (blockIdx.x * blockDim.x + threadIdx.x) >> 5;
  int lane   = threadIdx.x & 31;
  int nwarps = (gridDim.x * blockDim.x) >> 5;
  for (int e = gwarp; e < E; e += nwarps) {
    int s = src[e], d = dst[e];
    int h = lane >> 2;  // head for this lane's 4 columns
    float as = al_src[s * GAT_HEADS + h];
    float ad = al_dst[d * GAT_HEADS + h];
    float mx = m[d * GAT_HEADS + h];
    float dn = denom[d * GAT_HEADS + h];
    float ex = expf(leaky(as + ad) - mx);
    float alpha = ex / (dn == 0.0f ? 1.0f : dn);
    int col = lane * 4;
    float4 v = *(const float4*)(xp + s * GAT_OUT + col);
    float* o = out + d * GAT_OUT + col;
    atomicAdd(o + 0, v.x * alpha);
    atomicAdd(o + 1, v.y * alpha);
    atomicAdd(o + 2, v.z * alpha);
    atomicAdd(o + 3, v.w * alpha);
  }
}

// ---------------------------------------------------------------------------
// Kernel 7: out = ELU(out + bias), in place.
// ---------------------------------------------------------------------------
__global__ void gat_finalize(float* __restrict__ out,
                             const float* __restrict__ bias, int N) {
  int i = blockIdx.x * blockDim.x + threadIdx.x;
  if (i >= N * GAT_OUT) return;
  float v = out[i] + bias[i & 127];
  out[i] = v > 0.0f ? v : expm1f(v);
}

// ---------------------------------------------------------------------------
extern "C" void kernel_launch(void* const* d_in, const int* in_sizes, int n_in,
                              void* d_out, int out_size, void* d_ws,
                              size_t ws_size, hipStream_t stream) {
  const float* x      = (const float*)d_in[0];  // [N,128]
  const int*   ei     = (const int*)d_in[1];    // [2,E]
  const float* W      = (const float*)d_in[2];  // [128,8,16] == [128,128]
  const float* a_src  = (const float*)d_in[3];  // [8,16]
  const float* a_dst  = (const float*)d_in[4];  // [8,16]
  const float* bias   = (const float*)d_in[5];  // [128]

  const int N = in_sizes[0] / GAT_IN_DIM;
  const int E = in_sizes[1] / 2;
  const int* src = ei;
  const int* dst = ei + E;

  float* out = (float*)d_out;  // [N,128], used as f32 accumulator

  // Workspace layout (floats): xp[N*128] | al_src[N*8] | al_dst[N*8] |
  //                            m[N*8] | denom[N*8]
  float* ws     = (float*)d_ws;
  float* xp     = ws;
  float* al_src = xp + (size_t)N * GAT_OUT;
  float* al_dst = al_src + (size_t)N * GAT_HEADS;
  float* mbuf   = al_dst + (size_t)N * GAT_HEADS;
  float* denom  = mbuf + (size_t)N * GAT_HEADS;

  // 1) init out/m/denom
  {
    int total = N * GAT_OUT;
    gat_init<<<(total + 255) / 256, 256, 0, stream>>>(out, mbuf, denom, N);
  }
  // 2) projection GEMM (f32 WMMA)
  gat_proj_wmma<<<(N + 15) / 16, 256, 0, stream>>>(x, W, xp, N);
  // 3) per-node attention halves
  {
    int total = N * GAT_HEADS;
    gat_al<<<(total + 255) / 256, 256, 0, stream>>>(xp, a_src, a_dst, al_src,
                                                    al_dst, N);
  }
  // 4) segment max
  gat_edge_max<<<(E + 255) / 256, 256, 0, stream>>>(src, dst, al_src, al_dst,
                                                    mbuf, E);
  // 5) segment exp-sum
  gat_edge_expsum<<<(E + 255) / 256, 256, 0, stream>>>(src, dst, al_src,
                                                       al_dst, mbuf, denom, E);
  // 6) weighted scatter-aggregate (1 wave / edge, grid-stride)
  gat_edge_agg<<<8192, 256, 0, stream>>>(src, dst, al_src, al_dst, mbuf, denom,
                                         xp, out, E);
  // 7) bias + ELU
  {
    int total = N * GAT_OUT;
    gat_finalize<<<(total + 255) / 256, 256, 0, stream>>>(out, bias, N);
  }
}